// Latent_Space_86955907874939
// MI455X (gfx1250) — compile-verified
//
#include <hip/hip_runtime.h>

typedef __attribute__((ext_vector_type(2))) float v2f;
typedef __attribute__((ext_vector_type(8))) float v8f;

#define C_DIM   256
#define K_CODES 512
#define H_DIM   64
#define W_DIM   64
#define N_PER_B 4096
#define B_DIM   16
#define N_VEC   (B_DIM * N_PER_B)                           /* 65536 */
#define OUT_ELEMS ((size_t)B_DIM * C_DIM * H_DIM * W_DIM)   /* 16777216 */
#define HW      (H_DIM * W_DIM)                             /* 4096 */

// ---------------- kernel 1: codebook squared norms e2[k] ----------------
__global__ __launch_bounds__(32) void vq_e2_kernel(const float* __restrict__ cb,
                                                   float* __restrict__ e2) {
  const int code = blockIdx.x;
  const int lane = threadIdx.x;
  const float* row = cb + (size_t)code * C_DIM;
  float s = 0.f;
#pragma unroll
  for (int i = 0; i < C_DIM / 32; ++i) {
    float x = row[lane + 32 * i];
    s = __builtin_fmaf(x, x, s);
  }
#pragma unroll
  for (int off = 16; off > 0; off >>= 1) s += __shfl_xor(s, off, 32);
  if (lane == 0) e2[code] = s;
}

// ---------------- kernel 2: fp32 WMMA GEMM + argmin ----------------
// Grid: 1024 blocks of 128 threads (4 waves). Block tile: 64 vectors
// (fixed b; h in [h0,h0+2); w in [wbase,wbase+32)), all 512 codes, K=256.
__global__ __launch_bounds__(128) void vq_argmin_kernel(
    const float* __restrict__ pre,       // [16,256,64,64]
    const float* __restrict__ cb,        // [512,256]
    const float* __restrict__ e2,        // [512]
    int* __restrict__ idx_out)           // [65536]
{
  __shared__ float xlds[64 * C_DIM];     // 64 KB: xlds[v][c], v = wLoc*2 + hLoc
  const int t    = threadIdx.x;
  const int lane = t & 31;
  const int wave = t >> 5;

  const int blk   = blockIdx.x;
  const int b     = blk >> 6;            // 64 blocks per batch
  const int rem   = blk & 63;
  const int h0    = (rem & 31) * 2;
  const int wbase = (rem >> 5) * 32;
  const size_t bBase = (size_t)b * C_DIM * HW;

  // ---- stage X tile: 512 segments of 32 contiguous w-floats (coalesced 128B)
  for (int it = 0; it < 128; ++it) {
    int seg  = wave + 4 * it;            // 0..511
    int hLoc = seg & 1;
    int c    = seg >> 1;
    float val = pre[bBase + (size_t)c * HW + (size_t)(h0 + hLoc) * W_DIM + wbase + lane];
    xlds[(lane * 2 + hLoc) * C_DIM + c] = val;
  }
  __syncthreads();

  // ---- A strip into registers, in WMMA f32 16x16x4 A-operand layout:
  // lanes 0-15: row M=lane, K = 4*kc + {0,1}; lanes 16-31: K = 4*kc + {2,3}
  const int hi   = lane >> 4;            // 0/1
  const int mrow = lane & 15;
  const int vrow = wave * 16 + mrow;     // this lane's A row (local vector id)
  v2f ka[64];
  {
    const float* abase = &xlds[vrow * C_DIM + 2 * hi];
#pragma unroll
    for (int kc = 0; kc < 64; ++kc)
      ka[kc] = *(const v2f*)(abase + 4 * kc);   // ds_load_b64, 8B aligned
  }

  float runMin[8];
  int   runIdx[8];
#pragma unroll
  for (int r = 0; r < 8; ++r) { runMin[r] = 3.4e38f; runIdx[r] = 0; }

  for (int tile = 0; tile < 32; ++tile) {
    const int n0 = tile * 16;
    // B operand: lane holds column N = n0+mrow, K = 4*kc + 2*hi + {0,1}
    const float* bptr = cb + (size_t)(n0 + mrow) * C_DIM + 2 * hi;
    const float  e2v  = e2[n0 + mrow];
    v8f acc = {0.f, 0.f, 0.f, 0.f, 0.f, 0.f, 0.f, 0.f};
#pragma unroll
    for (int kc = 0; kc < 64; ++kc) {
      v2f bb = *(const v2f*)(bptr + 4 * kc);    // global_load_b64 (L0/L2 hit)
      acc = __builtin_amdgcn_wmma_f32_16x16x4_f32(
          false, ka[kc], false, bb, (short)0, acc, false, false);
    }
    // C layout: this lane's column is N = n0+mrow; VGPR r -> row M = 8*hi + r
#pragma unroll
    for (int r = 0; r < 8; ++r) {
      float s = __builtin_fmaf(-2.0f, acc[r], e2v);  // ||e||^2 - 2 x.e
      if (s < runMin[r]) { runMin[r] = s; runIdx[r] = n0 + mrow; }
    }
  }

  // ---- cross-lane argmin over the 16 columns held within each half-wave
#pragma unroll
  for (int r = 0; r < 8; ++r) {
    float v  = runMin[r];
    int   ix = runIdx[r];
#pragma unroll
    for (int off = 1; off < 16; off <<= 1) {
      float ov = __shfl_xor(v, off, 32);
      int   oi = __shfl_xor(ix, off, 32);
      if (ov < v || (ov == v && oi < ix)) { v = ov; ix = oi; }
    }
    if (mrow == 0) {
      int m    = 8 * hi + r;
      int vloc = wave * 16 + m;          // local vector id
      int wLoc = vloc >> 1, hLoc = vloc & 1;
      int n    = (wbase + wLoc) * H_DIM + (h0 + hLoc);  // n = w*H + h
      idx_out[b * N_PER_B + n] = ix;
    }
  }
}

// ---------------- kernel 3: gather codebook rows, write out, loss partials ----
// Grid: 2048 blocks of 256 threads; block = 32 consecutive n (fixed b, i).
__global__ __launch_bounds__(256) void vq_gather_kernel(
    const float* __restrict__ pre,
    const float* __restrict__ cb,
    const int*   __restrict__ idx,
    float* __restrict__ out,
    float* __restrict__ partials)
{
  __shared__ float red[256];
  const int t    = threadIdx.x;
  const int lane = t & 31;
  const int wv   = t >> 5;
  const int blk  = blockIdx.x;
  const int b    = blk >> 7;             // 128 blocks per batch
  const int n    = (blk & 127) * 32 + lane;
  const int i    = n >> 6;               // out row
  const int j    = n & 63;               // out col
  const size_t bBase = (size_t)b * C_DIM * HW;

  const int kv = idx[b * N_PER_B + n];
  const float* crow = cb + (size_t)kv * C_DIM;

  float accl = 0.f;
  for (int it = 0; it < 32; ++it) {
    int c = wv + 8 * it;                 // each wave: fixed c per iter
    float qo = crow[c];                  // gather, cache-resident codebook
    float qi = pre[bBase + (size_t)c * HW + j * W_DIM + i];   // pre[b,c,h=j,w=i]
    out[bBase + (size_t)c * HW + i * W_DIM + j] = qo;         // coalesced in j
    float d = qo - qi;
    accl = __builtin_fmaf(d, d, accl);
  }

  red[t] = accl;
  __syncthreads();
  for (int s = 128; s > 0; s >>= 1) {
    if (t < s) red[t] += red[t + s];
    __syncthreads();
  }
  if (t == 0) partials[blk] = red[0];
}

// ---------------- kernel 4: deterministic final loss reduction ----------------
__global__ __launch_bounds__(256) void vq_loss_kernel(const float* __restrict__ partials,
                                                      float* __restrict__ out_loss) {
  __shared__ float red[256];
  const int t = threadIdx.x;
  float s = 0.f;
  for (int i = t; i < 2048; i += 256) s += partials[i];   // fixed order
  red[t] = s;
  __syncthreads();
  for (int k = 128; k > 0; k >>= 1) {
    if (t < k) red[t] += red[t + k];
    __syncthreads();
  }
  if (t == 0)
    out_loss[0] = red[0] * 1.25f / 16777216.0f;  // (1 + 0.25) * mean over B*N*C
}

extern "C" void kernel_launch(void* const* d_in, const int* in_sizes, int n_in,
                              void* d_out, int out_size, void* d_ws, size_t ws_size,
                              hipStream_t stream) {
  const float* pre = (const float*)d_in[0];   // [16,256,64,64] fp32
  const float* cb  = (const float*)d_in[1];   // [512,256] fp32
  float* out = (float*)d_out;                 // 16777216 floats + 1 loss

  float* e2       = (float*)d_ws;                                   // 512 f
  int*   idxw     = (int*)((char*)d_ws + 512 * sizeof(float));      // 65536 i32
  float* partials = (float*)((char*)d_ws + 512 * sizeof(float)
                                         + (size_t)N_VEC * sizeof(int)); // 2048 f

  vq_e2_kernel    <<<K_CODES, 32, 0, stream>>>(cb, e2);
  vq_argmin_kernel<<<1024, 128, 0, stream>>>(pre, cb, e2, idxw);
  vq_gather_kernel<<<2048, 256, 0, stream>>>(pre, cb, idxw, out, partials);
  vq_loss_kernel  <<<1, 256, 0, stream>>>(partials, out + OUT_ELEMS);
}